// SEModule_44538810859952
// MI455X (gfx1250) — compile-verified
//
#include <hip/hip_runtime.h>
#include <math.h>

typedef __attribute__((ext_vector_type(2))) float v2f;
typedef __attribute__((ext_vector_type(4))) float v4f;
typedef __attribute__((ext_vector_type(8))) float v8f;

#define SE_B   16
#define SE_C   256
#define SE_CR  64
#define SE_HW  16384   // 128*128

// ---------------------------------------------------------------------------
// Kernel 1: global average pool. One block per (b,c) plane (64 KB contiguous).
// 256 threads x 16 float4 each, LDS tree reduction. Pure streaming read.
// ---------------------------------------------------------------------------
__global__ __launch_bounds__(256)
void se_pool(const float* __restrict__ x, float* __restrict__ s) {
  __shared__ float red[256];
  const int bc = blockIdx.x;
  const int t  = threadIdx.x;
  const v4f* xp = (const v4f*)(x + (size_t)bc * SE_HW);
  float sum = 0.f;
#pragma unroll
  for (int i = 0; i < 16; ++i) {
    v4f v = xp[i * 256 + t];
    sum += v.x + v.y + v.z + v.w;
  }
  red[t] = sum;
  __syncthreads();
  for (int off = 128; off > 0; off >>= 1) {
    if (t < off) red[t] += red[t + off];
    __syncthreads();
  }
  if (t == 0) s[bc] = red[0] * (1.0f / SE_HW);
}

// ---------------------------------------------------------------------------
// Kernel 2: excite MLP with f32 WMMA (V_WMMA_F32_16X16X4_F32).
// Single workgroup, 4 waves (wave32). M=16 == batch.
//   GEMM1: h[16,64]  = s[16,256] @ w1[64,256]^T  (+b1, leaky_relu) -> LDS
//   GEMM2: g[16,256] = h[16,64]  @ w2[256,64]^T  (+b2, sigmoid)    -> ws
// f32 A-frag (16x4): lanes 0-15 hold K=k0..k0+1, lanes 16-31 K=k0+2..k0+3,
//                    row M = lane%16. B addressed symmetrically (col N = lane%16).
// f32 D-frag (16x16): VGPR v, lane l -> M = v + 8*(l>=16), N = l%16.
// ---------------------------------------------------------------------------
__global__ __launch_bounds__(128)
void se_mlp(const float* __restrict__ s,  const float* __restrict__ w1,
            const float* __restrict__ b1, const float* __restrict__ w2,
            const float* __restrict__ b2, float* __restrict__ g) {
  __shared__ float h[SE_B][SE_CR];
  const int t    = threadIdx.x;
  const int wv   = t >> 5;        // wave 0..3
  const int lane = t & 31;
  const int half = lane >> 4;     // 0 or 1
  const int l16  = lane & 15;     // M for A-frag, N for B/D-frag
  const int koff = half * 2;      // K sub-offset within 4-wide step

  // ---- GEMM1: wave wv produces h columns [wv*16, wv*16+16) ----
  const int n1 = wv * 16 + l16;   // output column == w1 row
  v8f acc;
  {
    const float bias1 = b1[n1];
#pragma unroll
    for (int v = 0; v < 8; ++v) acc[v] = bias1;
  }
  for (int k0 = 0; k0 < SE_C; k0 += 4) {
    v2f a, b;
    a.x = s[l16 * SE_C + k0 + koff];
    a.y = s[l16 * SE_C + k0 + koff + 1];
    b.x = w1[n1 * SE_C + k0 + koff];
    b.y = w1[n1 * SE_C + k0 + koff + 1];
    acc = __builtin_amdgcn_wmma_f32_16x16x4_f32(false, a, false, b,
                                                (short)0, acc, false, false);
  }
#pragma unroll
  for (int v = 0; v < 8; ++v) {
    float hv = acc[v];
    hv = hv > 0.f ? hv : 0.2f * hv;                 // leaky_relu(0.2)
    h[v + 8 * half][wv * 16 + l16] = hv;
  }
  __syncthreads();

  // ---- GEMM2: wave wv produces g column tiles wv*4 .. wv*4+3 ----
  for (int tile = 0; tile < 4; ++tile) {
    const int n2 = (wv * 4 + tile) * 16 + l16;      // output channel == w2 row
    v8f acc2;
    {
      const float bias2 = b2[n2];
#pragma unroll
      for (int v = 0; v < 8; ++v) acc2[v] = bias2;
    }
#pragma unroll
    for (int k0 = 0; k0 < SE_CR; k0 += 4) {
      v2f a, b;
      a.x = h[l16][k0 + koff];
      a.y = h[l16][k0 + koff + 1];
      b.x = w2[n2 * SE_CR + k0 + koff];
      b.y = w2[n2 * SE_CR + k0 + koff + 1];
      acc2 = __builtin_amdgcn_wmma_f32_16x16x4_f32(false, a, false, b,
                                                   (short)0, acc2, false, false);
    }
#pragma unroll
    for (int v = 0; v < 8; ++v) {
      const float gv = 1.0f / (1.0f + __expf(-acc2[v]));   // sigmoid
      g[(v + 8 * half) * SE_C + n2] = gv;
    }
  }
}

// ---------------------------------------------------------------------------
// Kernel 3: out = x * g[b,c]. One block per plane; nontemporal (last-use /
// write-once) so streamed data doesn't evict useful L2 lines.
// ---------------------------------------------------------------------------
__global__ __launch_bounds__(256)
void se_scale(const float* __restrict__ x, const float* __restrict__ g,
              float* __restrict__ out) {
  const int bc = blockIdx.x;
  const int t  = threadIdx.x;
  const float gate = g[bc];
  const v4f* xp = (const v4f*)(x  + (size_t)bc * SE_HW);
  v4f*       op = (v4f*)      (out + (size_t)bc * SE_HW);
#pragma unroll
  for (int i = 0; i < 16; ++i) {
    v4f v = __builtin_nontemporal_load(&xp[i * 256 + t]);
    v *= gate;
    __builtin_nontemporal_store(v, &op[i * 256 + t]);
  }
}

extern "C" void kernel_launch(void* const* d_in, const int* in_sizes, int n_in,
                              void* d_out, int out_size, void* d_ws, size_t ws_size,
                              hipStream_t stream) {
  const float* x  = (const float*)d_in[0];   // [16,256,128,128]
  const float* w1 = (const float*)d_in[1];   // [64,256]
  const float* b1 = (const float*)d_in[2];   // [64]
  const float* w2 = (const float*)d_in[3];   // [256,64]
  const float* b2 = (const float*)d_in[4];   // [256]
  float* out = (float*)d_out;

  float* s = (float*)d_ws;                   // [16,256] pooled means
  float* g = s + SE_B * SE_C;                // [16,256] gates

  se_pool <<<SE_B * SE_C, 256, 0, stream>>>(x, s);
  se_mlp  <<<1, 128, 0, stream>>>(s, w1, b1, w2, b2, g);
  se_scale<<<SE_B * SE_C, 256, 0, stream>>>(x, g, out);
}